// GCN_46256797777943
// MI455X (gfx1250) — compile-verified
//
#include <hip/hip_runtime.h>

typedef float v2f __attribute__((ext_vector_type(2)));
typedef float v8f __attribute__((ext_vector_type(8)));

#define NNODES 50000
#define NEDGES 800000
#define HID 64
#define NGRAPHS 512
#define NCLASSES 10

// ---------------- degree / normalization ----------------

__global__ __launch_bounds__(256) void k_init_deg(float* __restrict__ deg, int n) {
    int i = blockIdx.x * blockDim.x + threadIdx.x;
    if (i < n) deg[i] = 1.0f;  // self-loop contributes 1
}

__global__ __launch_bounds__(256) void k_deg_accum(const int* __restrict__ dst,
                                                   float* __restrict__ deg, int e) {
    int i = blockIdx.x * blockDim.x + threadIdx.x;
    if (i < e) atomicAdd(&deg[dst[i]], 1.0f);
}

__global__ __launch_bounds__(256) void k_dinv(float* __restrict__ deg, int n) {
    int i = blockIdx.x * blockDim.x + threadIdx.x;
    if (i < n) deg[i] = rsqrtf(deg[i]);  // deg >= 1 always
}

// ---------------- GEMM: C[n x 64] = (relu?)A[n x 64] @ W[64 x 64] via WMMA fp32 ----------------

__global__ __launch_bounds__(256)
void k_gemm_wmma(const float* __restrict__ A, const float* __restrict__ W,
                 float* __restrict__ C, int n, int relu_in) {
    __shared__ float sW[64 * 64];
    const int tid = threadIdx.x;
    #pragma unroll
    for (int i = 0; i < 16; ++i) sW[tid + i * 256] = W[tid + i * 256];
    __syncthreads();

    const int wave = tid >> 5;
    const int lane = tid & 31;
    const int half = lane >> 4;   // 0 or 1
    const int l16  = lane & 15;
    const int m0   = (blockIdx.x * 8 + wave) * 16;
    if (m0 >= n) return;  // wave-uniform

    v8f c0 = {}, c1 = {}, c2 = {}, c3 = {};

    const int arow = m0 + l16;                 // A-frag: M = lane&15
    const float* __restrict__ Arow = A + (size_t)arow * 64;

    #pragma unroll
    for (int k0 = 0; k0 < 64; k0 += 4) {
        const int kb = k0 + half * 2;          // A: VGPR0 holds K=kb, VGPR1 K=kb+1
        float a0 = Arow[kb];
        float a1 = Arow[kb + 1];
        if (relu_in) { a0 = fmaxf(a0, 0.0f); a1 = fmaxf(a1, 0.0f); }
        v2f a = {a0, a1};

        // B frag: VGPR v, half h -> K = k0 + v + 2h, N = l16 (+ 16*ntile)
        const float* __restrict__ Wk0 = sW + (size_t)kb * 64;        // row K=kb
        const float* __restrict__ Wk1 = sW + (size_t)(kb + 1) * 64;  // row K=kb+1
        v2f b0 = {Wk0[l16 +  0], Wk1[l16 +  0]};
        v2f b1 = {Wk0[l16 + 16], Wk1[l16 + 16]};
        v2f b2 = {Wk0[l16 + 32], Wk1[l16 + 32]};
        v2f b3 = {Wk0[l16 + 48], Wk1[l16 + 48]};

        c0 = __builtin_amdgcn_wmma_f32_16x16x4_f32(false, a, false, b0, (short)0, c0, false, false);
        c1 = __builtin_amdgcn_wmma_f32_16x16x4_f32(false, a, false, b1, (short)0, c1, false, false);
        c2 = __builtin_amdgcn_wmma_f32_16x16x4_f32(false, a, false, b2, (short)0, c2, false, false);
        c3 = __builtin_amdgcn_wmma_f32_16x16x4_f32(false, a, false, b3, (short)0, c3, false, false);
    }

    // C/D layout: VGPR r -> M = m0 + r + 8*half, N = l16 (+16*ntile)
    #pragma unroll
    for (int r = 0; r < 8; ++r) {
        const int row = m0 + half * 8 + r;
        if (row < n) {
            float* __restrict__ Cr = C + (size_t)row * 64 + l16;
            Cr[0]  = c0[r];
            Cr[16] = c1[r];
            Cr[32] = c2[r];
            Cr[48] = c3[r];
        }
    }
}

// ---------------- aggregation ----------------

// a[i,f] = h[i,f] * dinv[i]^2 + bias[f]   (self-loop + bias init)
__global__ __launch_bounds__(256)
void k_selfbias(const float* __restrict__ h, const float* __restrict__ dinv,
                const float* __restrict__ bias, float* __restrict__ a, int n) {
    int t = blockIdx.x * blockDim.x + threadIdx.x;
    if (t < n * HID) {
        int i = t >> 6;
        int f = t & 63;
        float di = dinv[i];
        a[t] = h[t] * di * di + bias[f];
    }
}

// a[dst,f] += h[src,f] * dinv[src] * dinv[dst]   (4 edges per block, 64 threads/edge)
__global__ __launch_bounds__(256)
void k_edge_scatter(const float* __restrict__ h, const float* __restrict__ dinv,
                    const int* __restrict__ src, const int* __restrict__ dst,
                    float* __restrict__ a, int e) {
    int t = blockIdx.x * blockDim.x + threadIdx.x;
    int ei = t >> 6;
    int f = t & 63;
    if (ei < e) {
        int s = src[ei];
        int d = dst[ei];
        float nrm = dinv[s] * dinv[d];
        atomicAdd(&a[(size_t)d * 64 + f], h[(size_t)s * 64 + f] * nrm);
    }
}

// ---------------- pooling + head ----------------

__global__ __launch_bounds__(256) void k_zero(float* __restrict__ p, int n) {
    int i = blockIdx.x * blockDim.x + threadIdx.x;
    if (i < n) p[i] = 0.0f;
}

// sums[g,f] += relu(a[i,f]); cnt[g] += 1 (once per node)
__global__ __launch_bounds__(256)
void k_pool(const float* __restrict__ a, const int* __restrict__ batch,
            float* __restrict__ sums, float* __restrict__ cnt, int n) {
    int t = blockIdx.x * blockDim.x + threadIdx.x;
    if (t < n * HID) {
        int i = t >> 6;
        int f = t & 63;
        int g = batch[i];
        atomicAdd(&sums[(size_t)g * HID + f], fmaxf(a[t], 0.0f));
        if (f == 0) atomicAdd(&cnt[g], 1.0f);
    }
}

// out[g,c] = (1/max(cnt,1)) * dot(sums[g,:], Wl[:,c]) + bl[c]
__global__ __launch_bounds__(256)
void k_head(const float* __restrict__ sums, const float* __restrict__ cnt,
            const float* __restrict__ Wl, const float* __restrict__ bl,
            float* __restrict__ out) {
    int t = blockIdx.x * blockDim.x + threadIdx.x;
    if (t < NGRAPHS * NCLASSES) {
        int g = t / NCLASSES;
        int c = t % NCLASSES;
        float inv = 1.0f / fmaxf(cnt[g], 1.0f);
        float acc = 0.0f;
        const float* sg = sums + (size_t)g * HID;
        #pragma unroll
        for (int k = 0; k < HID; ++k) acc += sg[k] * Wl[k * NCLASSES + c];
        out[t] = acc * inv + bl[c];
    }
}

// ---------------- launch ----------------

extern "C" void kernel_launch(void* const* d_in, const int* in_sizes, int n_in,
                              void* d_out, int out_size, void* d_ws, size_t ws_size,
                              hipStream_t stream) {
    const float* x  = (const float*)d_in[0];
    const float* W1 = (const float*)d_in[1];
    const float* b1 = (const float*)d_in[2];
    const float* W2 = (const float*)d_in[3];
    const float* b2 = (const float*)d_in[4];
    const float* Wl = (const float*)d_in[5];
    const float* bl = (const float*)d_in[6];
    const int* edge  = (const int*)d_in[7];
    const int* batch = (const int*)d_in[8];
    const int* src = edge;
    const int* dst = edge + NEDGES;
    float* out = (float*)d_out;

    float* ws   = (float*)d_ws;
    float* h    = ws;                               // [NNODES,64]
    float* a    = h + (size_t)NNODES * HID;         // [NNODES,64]
    float* dinv = a + (size_t)NNODES * HID;         // [NNODES]
    float* sums = dinv + NNODES;                    // [NGRAPHS,64]
    float* cnt  = sums + (size_t)NGRAPHS * HID;     // [NGRAPHS]

    const int TPB = 256;
    const int gN   = (NNODES + TPB - 1) / TPB;
    const int gE   = (NEDGES + TPB - 1) / TPB;
    const int gNF  = (NNODES * HID + TPB - 1) / TPB;
    const int gEd  = (NEDGES + 3) / 4;              // 4 edges per block
    const int gGmm = ((NNODES + 15) / 16 + 7) / 8;  // 8 row-tiles per block
    const int gPZ  = (NGRAPHS * HID + NGRAPHS + TPB - 1) / TPB;
    const int gHd  = (NGRAPHS * NCLASSES + TPB - 1) / TPB;

    // normalization
    k_init_deg<<<gN, TPB, 0, stream>>>(dinv, NNODES);
    k_deg_accum<<<gE, TPB, 0, stream>>>(dst, dinv, NEDGES);
    k_dinv<<<gN, TPB, 0, stream>>>(dinv, NNODES);

    // layer 1
    k_gemm_wmma<<<gGmm, TPB, 0, stream>>>(x, W1, h, NNODES, 0);
    k_selfbias<<<gNF, TPB, 0, stream>>>(h, dinv, b1, a, NNODES);
    k_edge_scatter<<<gEd, TPB, 0, stream>>>(h, dinv, src, dst, a, NEDGES);

    // layer 2 (ReLU of layer-1 output fused into GEMM A-load)
    k_gemm_wmma<<<gGmm, TPB, 0, stream>>>(a, W2, h, NNODES, 1);
    k_selfbias<<<gNF, TPB, 0, stream>>>(h, dinv, b2, a, NNODES);
    k_edge_scatter<<<gEd, TPB, 0, stream>>>(h, dinv, src, dst, a, NEDGES);

    // pooling (layer-2 ReLU fused) + head
    k_zero<<<gPZ, TPB, 0, stream>>>(sums, NGRAPHS * HID + NGRAPHS);
    k_pool<<<gNF, TPB, 0, stream>>>(a, batch, sums, cnt, NNODES);
    k_head<<<gHd, TPB, 0, stream>>>(sums, cnt, Wl, bl, out);
}